// RiemannFMDualStreamCross_76776835383328
// MI455X (gfx1250) — compile-verified
//
#include <hip/hip_runtime.h>
#include <hip/hip_bf16.h>

#define B_  4
#define N_  256
#define ND  256
#define ED  128
#define PW  132   // LDS pitch for W4 (floats), padded vs 64 banks
#define PA  132   // LDS pitch for pair tile

typedef __attribute__((ext_vector_type(2))) float v2f;
typedef __attribute__((ext_vector_type(8))) float v8f;

__device__ __forceinline__ float silu_f(float x) {
    return x * (1.0f / (1.0f + __expf(-x)));
}

// ---------------------------------------------------------------------------
// Kernel 1: aproj = h @ W3a, bproj = h @ W3b   (tiny: 2 x 67 MFLOP)
// grid = B*N blocks, 128 threads
// ---------------------------------------------------------------------------
__global__ void proj_kernel(const float* __restrict__ h,
                            const float* __restrict__ W3a,
                            const float* __restrict__ W3b,
                            float* __restrict__ aproj,
                            float* __restrict__ bproj) {
    __shared__ float sH[ND];
    const int row = blockIdx.x;           // b*N + n
    const int t   = threadIdx.x;          // 0..127 -> output channel
    sH[t]       = h[(size_t)row * ND + t];
    sH[t + 128] = h[(size_t)row * ND + t + 128];
    __syncthreads();
    float da = 0.f, db = 0.f;
    #pragma unroll 8
    for (int k = 0; k < ND; ++k) {
        const float hv = sH[k];
        da = fmaf(hv, W3a[k * ED + t], da);
        db = fmaf(hv, W3b[k * ED + t], db);
    }
    aproj[(size_t)row * ED + t] = da;
    bproj[(size_t)row * ED + t] = db;
}

// ---------------------------------------------------------------------------
// Kernel 2: fused pass over g.
//  per block: (b, i, j0 in {0,128}); 256 threads = 8 waves.
//   pair[m,c] = silu(aproj[i,c] + bproj[j0+m,c] + b3[c])   -> LDS
//   acc      = pair(128x128) @ W4(128x128) via V_WMMA_F32_16X16X4_F32
//   g_out    = g + acc + b4                 (residual, g streamed once, NT)
//   partial  = sum_m mask[j0+m] * g[..]     (deterministic, no atomics)
// ---------------------------------------------------------------------------
__global__ void __launch_bounds__(256)
edge_kernel(const float* __restrict__ g,
            const int*   __restrict__ mask,
            const float* __restrict__ aproj,
            const float* __restrict__ bproj,
            const float* __restrict__ b3,
            const float* __restrict__ W4,
            const float* __restrict__ b4,
            float* __restrict__ g_out,
            float* __restrict__ partial) {
    extern __shared__ float smem[];
    float* sW4   = smem;                  // 128*PW
    float* sA    = sW4 + 128 * PW;        // 128*PA
    float* arow  = sA + 128 * PA;         // 128  (aproj row + b3)
    float* sMask = arow + 128;            // 128  (mask[j0..j0+127] as float)
    float* sRed  = sMask + 128;           // 256  (cross-half reduction)

    const int bid  = blockIdx.x;
    const int part = bid & 1;
    const int i    = (bid >> 1) & (N_ - 1);
    const int b    = bid >> 9;
    const int j0   = part * 128;
    const int t    = threadIdx.x;

    // stage W4 into LDS (b128 loads/stores)
    {
        const float4* w44 = (const float4*)W4;
        for (int idx = t; idx < (ED * ED) / 4; idx += 256) {
            const int k = idx >> 5, c4 = idx & 31;
            const float4 v = w44[idx];
            *(float4*)&sW4[k * PW + c4 * 4] = v;
        }
    }
    if (t < 128) {
        arow[t]  = aproj[((size_t)(b * N_ + i)) * ED + t] + b3[t];
        sMask[t] = mask[b * N_ + j0 + t] ? 1.0f : 0.0f;
    }
    __syncthreads();

    // pair activations into LDS (b128 path)
    {
        const float4* bp4 = (const float4*)(bproj + ((size_t)(b * N_) + j0) * ED);
        for (int idx = t; idx < 128 * (ED / 4); idx += 256) {
            const int m = idx >> 5, c4 = idx & 31, c = c4 * 4;
            const float4 bv = bp4[idx];
            const float4 av = *(const float4*)&arow[c];
            float4 o;
            o.x = silu_f(av.x + bv.x);
            o.y = silu_f(av.y + bv.y);
            o.z = silu_f(av.z + bv.z);
            o.w = silu_f(av.w + bv.w);
            *(float4*)&sA[m * PA + c] = o;
        }
    }
    __syncthreads();

    const int w  = t >> 5;     // wave id -> output-channel slab n0 = 16*w
    const int l  = t & 31;
    const int hl = l >> 4;     // lane half (K pair / M+8 selector)
    const int ln = l & 15;
    const int n0 = w * 16;

    v8f acc[8];
    const v8f vz = {0.f, 0.f, 0.f, 0.f, 0.f, 0.f, 0.f, 0.f};
    #pragma unroll
    for (int ms = 0; ms < 8; ++ms) acc[ms] = vz;

    // K = 128 in 32 steps of 4. Batched fragment loads per step + full
    // unroll so the scheduler can pipeline DS loads under the WMMAs.
    const float* aCol = sA + ln * PA;        // + ms*16*PA + kk
    const float* bCol = sW4 + n0 + ln;       // + kk*PW
    #pragma unroll
    for (int ks = 0; ks < 32; ++ks) {
        const int kk = ks * 4 + hl * 2;
        v2f bf = { bCol[kk * PW], bCol[(kk + 1) * PW] };
        v2f af[8];
        #pragma unroll
        for (int ms = 0; ms < 8; ++ms)
            af[ms] = *(const v2f*)(aCol + ms * 16 * PA + kk);
        #pragma unroll
        for (int ms = 0; ms < 8; ++ms)
            acc[ms] = __builtin_amdgcn_wmma_f32_16x16x4_f32(
                false, af[ms], false, bf, (short)0, acc[ms], false, false);
    }

    // residual store + masked sender-sum (single streamed read of g)
    const float  b4c   = b4[n0 + ln];
    const size_t gbase = ((size_t)(b * N_ + i)) * N_ * ED + (size_t)j0 * ED
                         + (n0 + ln);
    float jsum = 0.f;
    #pragma unroll
    for (int ms = 0; ms < 8; ++ms) {
        float gv[8];
        #pragma unroll
        for (int r = 0; r < 8; ++r) {
            const int m = ms * 16 + r + hl * 8;   // C/D layout: vgpr r -> M=r / r+8
            gv[r] = __builtin_nontemporal_load(&g[gbase + (size_t)m * ED]);
        }
        #pragma unroll
        for (int r = 0; r < 8; ++r) {
            const int m = ms * 16 + r + hl * 8;
            jsum = fmaf(gv[r], sMask[m], jsum);
            __builtin_nontemporal_store(gv[r] + acc[ms][r] + b4c,
                                        &g_out[gbase + (size_t)m * ED]);
        }
    }
    sRed[w * 32 + l] = jsum;
    __syncthreads();
    if (t < 128) {   // channel c = 16*w2 + l2; combine lane halves
        const int w2 = t >> 4, l2 = t & 15;
        const float v = sRed[w2 * 32 + l2] + sRed[w2 * 32 + l2 + 16];
        partial[((size_t)part * (B_ * N_) + b * N_ + i) * ED + t] = v;
    }
}

// ---------------------------------------------------------------------------
// Kernel 3: node update. grid = B*N blocks, 256 threads.
//  node_agg = mask_i * (partial0+partial1) / clip(count,1); MLP; h residual.
// ---------------------------------------------------------------------------
__global__ void node_kernel(const float* __restrict__ h,
                            const int*   __restrict__ mask,
                            const float* __restrict__ partial,
                            const float* __restrict__ W1,
                            const float* __restrict__ b1,
                            const float* __restrict__ W2,
                            const float* __restrict__ b2,
                            float* __restrict__ h_out) {
    __shared__ float sNA[ED];
    __shared__ float sT[ND];
    __shared__ int   sM[N_];
    const int row = blockIdx.x;        // b*N + i
    const int b   = row >> 8;
    const int i   = row & (N_ - 1);
    const int t   = threadIdx.x;       // 0..255
    sM[t] = mask[b * N_ + t] ? 1 : 0;
    __syncthreads();
    if (t < ED) {
        int cnt = 0;
        #pragma unroll 8
        for (int j = 0; j < N_; ++j) cnt += sM[j];
        const float mi = (float)sM[i];
        const float cf = fmaxf(mi * (float)cnt, 1.0f);
        const size_t off = (size_t)row * ED + t;
        const float s = partial[off] + partial[(size_t)(B_ * N_) * ED + off];
        sNA[t] = mi * s / cf;
    }
    __syncthreads();
    float a1 = b1[t];
    #pragma unroll 8
    for (int c = 0; c < ED; ++c) a1 = fmaf(sNA[c], W1[c * ND + t], a1);
    sT[t] = silu_f(a1);
    __syncthreads();
    float a2 = b2[t];
    #pragma unroll 8
    for (int k = 0; k < ND; ++k) a2 = fmaf(sT[k], W2[k * ND + t], a2);
    h_out[(size_t)row * ND + t] = h[(size_t)row * ND + t] + a2;
}

// ---------------------------------------------------------------------------
extern "C" void kernel_launch(void* const* d_in, const int* in_sizes, int n_in,
                              void* d_out, int out_size, void* d_ws, size_t ws_size,
                              hipStream_t stream) {
    const float* h   = (const float*)d_in[0];
    const float* g   = (const float*)d_in[1];
    const int*   nm  = (const int*)d_in[2];
    const float* W1  = (const float*)d_in[3];
    const float* b1  = (const float*)d_in[4];
    const float* W2  = (const float*)d_in[5];
    const float* b2  = (const float*)d_in[6];
    const float* W3a = (const float*)d_in[7];
    const float* W3b = (const float*)d_in[8];
    const float* b3  = (const float*)d_in[9];
    const float* W4  = (const float*)d_in[10];
    const float* b4  = (const float*)d_in[11];

    float* out_h = (float*)d_out;                          // (B,N,ND)
    float* out_g = out_h + (size_t)B_ * N_ * ND;           // (B,N,N,ED)

    float* aproj   = (float*)d_ws;                         // B*N*ED
    float* bproj   = aproj + (size_t)B_ * N_ * ED;         // B*N*ED
    float* partial = bproj + (size_t)B_ * N_ * ED;         // 2*B*N*ED

    proj_kernel<<<B_ * N_, 128, 0, stream>>>(h, W3a, W3b, aproj, bproj);

    const size_t shmem =
        (size_t)(128 * PW + 128 * PA + 128 + 128 + 256) * sizeof(float);
    edge_kernel<<<B_ * N_ * 2, 256, shmem, stream>>>(g, nm, aproj, bproj, b3,
                                                     W4, b4, out_g, partial);

    node_kernel<<<B_ * N_, 256, 0, stream>>>(h, nm, partial, W1, b1, W2, b2,
                                             out_h);
}